// GraphEncoder_70738111365179
// MI455X (gfx1250) — compile-verified
//
#include <hip/hip_runtime.h>

typedef __attribute__((ext_vector_type(2))) float v2f;
typedef __attribute__((ext_vector_type(8))) float v8f;

#define E_NODES    64
#define D_DIM      128
#define H_DIM      128
#define TOK_TILE   16      // tokens (GEMM rows) per block
#define LDS_STRIDE 132     // 128 + 4 pad -> row r starts at bank 4r

__launch_bounds__(256, 2)
__global__ void graph_encoder_kernel(const int*   __restrict__ ids,   // [TOK, 64]
                                     const float* __restrict__ emb,   // [V, 128]
                                     const float* __restrict__ W0,    // [128, 128]
                                     const float* __restrict__ b0,    // [128]
                                     float*       __restrict__ out)   // [TOK, 128]
{
    __shared__ float S[TOK_TILE * LDS_STRIDE];

    const int lane = threadIdx.x & 31;
    const int wave = threadIdx.x >> 5;          // 0..7
    const int t0   = blockIdx.x * TOK_TILE;     // first token of this block

    // ---------- Stage 1: gather + star-row sum ----------
    // center_out = (sum over all 64 neighbor embeddings) @ W0 + b0
    // Each wave accumulates 2 tokens; 32 lanes cover D=128 as float4.
    const float4* emb4 = (const float4*)emb;
    #pragma unroll
    for (int ti = 0; ti < 2; ++ti) {
        const int tl  = wave * 2 + ti;                 // local token 0..15
        const int tok = t0 + tl;
        const int* idp = ids + (size_t)tok * E_NODES;
        const int id_a = idp[lane];                    // neighbor  lane
        const int id_b = idp[lane + 32];               // neighbor  lane+32

        float ax = 0.f, ay = 0.f, az = 0.f, aw = 0.f;
        #pragma unroll
        for (int nb = 0; nb < 32; ++nb) {
            const int ia = __shfl(id_a, nb, 32);       // uniform broadcast
            const int ib = __shfl(id_b, nb, 32);
            const float4 va = emb4[(size_t)ia * (D_DIM / 4) + lane];
            const float4 vb = emb4[(size_t)ib * (D_DIM / 4) + lane];
            ax += va.x + vb.x;
            ay += va.y + vb.y;
            az += va.z + vb.z;
            aw += va.w + vb.w;
        }
        float4* dst = (float4*)&S[tl * LDS_STRIDE + lane * 4];
        *dst = make_float4(ax, ay, az, aw);
    }
    __syncthreads();

    // ---------- Stage 2: out[16,128] = S[16,128] @ W0[128,128] + b0 ----------
    // V_WMMA_F32_16X16X4_F32, K accumulated in 32 steps of 4.
    // A 16x4 layout:  lanes 0-15 = rows, VGPR0/1 hold K=0/1 ; lanes 16-31 hold K=2/3
    // B 4x16 layout:  lanes 0-15 = cols with K=0/1 ; lanes 16-31 = cols with K=2/3
    const int n0   = wave * 16;                 // this wave's 16-column tile of H
    const int row  = lane & 15;                 // A row / B-D column index
    const int koff = (lane < 16) ? 0 : 2;
    const int col  = n0 + row;

    v8f c = {};
    #pragma unroll
    for (int k = 0; k < 32; ++k) {
        const int kk = 4 * k + koff;
        v2f a = *(const v2f*)&S[row * LDS_STRIDE + kk];   // 8B aligned LDS load
        v2f b;
        b.x = W0[(size_t)kk * H_DIM + col];
        b.y = W0[(size_t)(kk + 1) * H_DIM + col];
        c = __builtin_amdgcn_wmma_f32_16x16x4_f32(
                /*neg_a=*/false, a, /*neg_b=*/false, b,
                /*c_mod=*/(short)0, c, /*reuse_a=*/false, /*reuse_b=*/false);
    }

    // D layout: VGPR r -> M=r (lanes 0-15) / M=r+8 (lanes 16-31), N=col
    const float bias  = b0[col];
    const int   rbase = (lane < 16) ? 0 : 8;
    #pragma unroll
    for (int r = 0; r < 8; ++r) {
        out[(size_t)(t0 + rbase + r) * H_DIM + col] = c[r] + bias;
    }
}

extern "C" void kernel_launch(void* const* d_in, const int* in_sizes, int n_in,
                              void* d_out, int out_size, void* d_ws, size_t ws_size,
                              hipStream_t stream) {
    const int*   ids = (const int*)  d_in[0];   // neighbor_ids [N,L,64] (int32)
    const float* emb = (const float*)d_in[1];   // emb_table    [V,128]
    const float* W0  = (const float*)d_in[2];   // [128,128]
    const float* b0  = (const float*)d_in[3];   // [128]
    float*       out = (float*)d_out;           // [N,L,128]

    const int n_tok  = in_sizes[0] / E_NODES;   // N*L = 8192
    const int blocks = n_tok / TOK_TILE;        // 512

    graph_encoder_kernel<<<blocks, 256, 0, stream>>>(ids, emb, W0, b0, out);
}